// GeoHead_61993557951043
// MI455X (gfx1250) — compile-verified
//
#include <hip/hip_runtime.h>
#include <hip/hip_bf16.h>
#include <math.h>

// Sinkhorn (SuperGlue optimal transport) for B=8, M=N=2048, 30 iterations.
// Strategy: keep the 134MB score matrix L2-resident (192MB L2 on MI455X).
// Row pass: one block/row, row staged into LDS via the CDNA5 async engine.
// Col pass: 32 cols/block, 64x32 tiles double-buffered through LDS via the
// async engine (single global read per element per iteration), chunked
// two-pass logsumexp merged into running (m,s) state.

#define BB 8
#define MM 2048
#define NN 2048
#define MP 2049
#define NP 2049
#define NEG_INF (-1.0e12f)
#define NUM_ITERS 30

__device__ __forceinline__ float block_reduce_max(float val, float* red) {
    int t = threadIdx.x;
    red[t] = val;
    __syncthreads();
    for (int s = 128; s > 0; s >>= 1) {
        if (t < s) red[t] = fmaxf(red[t], red[t + s]);
        __syncthreads();
    }
    float r = red[0];
    __syncthreads();
    return r;
}

__device__ __forceinline__ float block_reduce_sum(float val, float* red) {
    int t = threadIdx.x;
    red[t] = val;
    __syncthreads();
    for (int s = 128; s > 0; s >>= 1) {
        if (t < s) red[t] += red[t + s];
        __syncthreads();
    }
    float r = red[0];
    __syncthreads();
    return r;
}

// ---------------------------------------------------------------------------
// init: per-batch mask counts, norm, log_mu, log_nu, u=v=0
// ---------------------------------------------------------------------------
__global__ __launch_bounds__(256) void sink_init(
    const unsigned char* __restrict__ rowm, const unsigned char* __restrict__ colm,
    const float* __restrict__ alpha_p,
    float* __restrict__ u, float* __restrict__ v,
    float* __restrict__ log_mu, float* __restrict__ log_nu,
    float* __restrict__ normb) {
    __shared__ float red[256];
    int b = blockIdx.x;
    int t = threadIdx.x;

    float cr = 0.0f;
    for (int i = t; i < MM; i += 256) cr += (rowm[b * MM + i] != 0) ? 1.0f : 0.0f;
    float nvr = block_reduce_sum(cr, red);

    float cc = 0.0f;
    for (int j = t; j < NN; j += 256) cc += (colm[b * NN + j] != 0) ? 1.0f : 0.0f;
    float nvc = block_reduce_sum(cc, red);

    float nrm = -logf(nvr + nvc);
    if (t == 0) normb[b] = nrm;
    float lmu_bin = logf(nvc) + nrm;
    float lnu_bin = logf(nvr) + nrm;

    for (int i = t; i < MP; i += 256) {
        float lm = (i < MM) ? ((rowm[b * MM + i] != 0) ? nrm : NEG_INF) : lmu_bin;
        log_mu[b * MP + i] = lm;
        u[b * MP + i] = 0.0f;
    }
    for (int j = t; j < NP; j += 256) {
        float ln = (j < NN) ? ((colm[b * NN + j] != 0) ? nrm : NEG_INF) : lnu_bin;
        log_nu[b * NP + j] = ln;
        v[b * NP + j] = 0.0f;
    }
}

// ---------------------------------------------------------------------------
// row pass: one block per (b, row).  u = log_mu - LSE_j(S + v)
// Stages the 8KB score row into LDS with the CDNA5 async copy engine.
// ---------------------------------------------------------------------------
__global__ __launch_bounds__(256) void sink_row(
    const float* __restrict__ scores, const float* __restrict__ alpha_p,
    const unsigned char* __restrict__ rowm, const unsigned char* __restrict__ colm,
    const float* __restrict__ v, float* __restrict__ u,
    const float* __restrict__ log_mu) {
    __shared__ __align__(16) float srow[NN];   // 8KB staged score row
    __shared__ float red[256];

    int bi = blockIdx.x;
    int b = bi / MP;
    int i = bi % MP;
    int t = threadIdx.x;
    float alpha = alpha_p[0];
    const float* vb = v + b * NP;

    float xs[8];
    float xd;

    if (i < MM) {
        const float* rowp = scores + ((size_t)(b * MM + i)) * NN;
        // async stage: each lane moves 16B per issue, 2 issues -> 2048 floats
        #pragma unroll
        for (int c = 0; c < 2; ++c) {
            int e = (c * 256 + t) * 4;
            unsigned int lds_off = (unsigned int)(size_t)(&srow[e]);
            const float* g = rowp + e;
            asm volatile("global_load_async_to_lds_b128 %0, %1, off"
                         :: "v"(lds_off), "v"(g) : "memory");
        }
        asm volatile("s_wait_asynccnt 0" ::: "memory");
        __syncthreads();

        bool rowv = rowm[b * MM + i] != 0;
        #pragma unroll
        for (int k = 0; k < 8; ++k) {
            int j = t + 256 * k;
            bool colv = colm[b * NN + j] != 0;
            float s = (rowv && colv) ? srow[j] : NEG_INF;
            xs[k] = s + vb[j];
        }
        xd = (rowv ? alpha : NEG_INF) + vb[NN];
    } else {
        // dustbin row: S = alpha (masked by col validity), corner = alpha
        #pragma unroll
        for (int k = 0; k < 8; ++k) {
            int j = t + 256 * k;
            bool colv = colm[b * NN + j] != 0;
            xs[k] = (colv ? alpha : NEG_INF) + vb[j];
        }
        xd = alpha + vb[NN];
    }

    float m = -3.0e38f;
    #pragma unroll
    for (int k = 0; k < 8; ++k) m = fmaxf(m, xs[k]);
    if (t == 0) m = fmaxf(m, xd);
    float mr = block_reduce_max(m, red);

    float sm = 0.0f;
    #pragma unroll
    for (int k = 0; k < 8; ++k) sm += expf(xs[k] - mr);
    if (t == 0) sm += expf(xd - mr);
    float tot = block_reduce_sum(sm, red);

    if (t == 0) u[b * MP + i] = log_mu[b * MP + i] - (mr + logf(tot));
}

// ---------------------------------------------------------------------------
// col pass: block = 32 columns x 8-way row split.  v = log_nu - LSE_i(S + u)
// 64x32 score tiles double-buffered through LDS via the async engine.
// ---------------------------------------------------------------------------
#define CPB 32                  // columns per block
#define RPT 64                  // rows per tile
#define NTILES (MM / RPT)       // 32 tiles cover rows [0, 2048)
#define NJB ((NP + CPB - 1) / CPB)   // 65 column-groups per batch

__global__ __launch_bounds__(256) void sink_col(
    const float* __restrict__ scores, const float* __restrict__ alpha_p,
    const unsigned char* __restrict__ rowm, const unsigned char* __restrict__ colm,
    const float* __restrict__ u, float* __restrict__ v,
    const float* __restrict__ log_nu) {
    __shared__ __align__(16) float tile[2][RPT * CPB];  // 2 x 8KB
    __shared__ float red_m[256];
    __shared__ float red_s[256];

    int b  = blockIdx.x / NJB;
    int jb = blockIdx.x % NJB;
    int tt = threadIdx.x;
    int jx = tt & 31;
    int ty = tt >> 5;
    int j  = jb * CPB + jx;
    float alpha = alpha_p[0];
    const float* ub = u + b * MP;

    float m = -3.0e38f;   // running max
    float s = 0.0f;       // running sum (relative to m)

    if ((jb + 1) * CPB <= NN) {
        // ---- full path: all 32 columns are real score columns ----
        bool colv = colm[b * NN + j] != 0;
        const float* gbase = scores + (size_t)b * MM * NN + (size_t)jb * CPB;

        // stage tile 0: 512 b128 segments; seg -> row seg>>3, col-group seg&7
        #pragma unroll
        for (int c = 0; c < 2; ++c) {
            int seg = tt + 256 * c;
            int r = seg >> 3, g = seg & 7;
            unsigned int lo = (unsigned int)(size_t)(&tile[0][r * CPB + g * 4]);
            const float* gp = gbase + (size_t)r * NN + g * 4;
            asm volatile("global_load_async_to_lds_b128 %0, %1, off"
                         :: "v"(lo), "v"(gp) : "memory");
        }

        for (int ti = 0; ti < NTILES; ++ti) {
            const float* cur = tile[ti & 1];
            if (ti + 1 < NTILES) {
                // stage next tile into the other buffer (freed by the barrier
                // at the end of iteration ti-1), then wait only for tile ti.
                int r0n = (ti + 1) * RPT;
                float* nb = tile[(ti + 1) & 1];
                #pragma unroll
                for (int c = 0; c < 2; ++c) {
                    int seg = tt + 256 * c;
                    int r = seg >> 3, g = seg & 7;
                    unsigned int lo = (unsigned int)(size_t)(&nb[r * CPB + g * 4]);
                    const float* gp = gbase + (size_t)(r0n + r) * NN + g * 4;
                    asm volatile("global_load_async_to_lds_b128 %0, %1, off"
                                 :: "v"(lo), "v"(gp) : "memory");
                }
                asm volatile("s_wait_asynccnt 2" ::: "memory");
            } else {
                asm volatile("s_wait_asynccnt 0" ::: "memory");
            }
            __syncthreads();

            // consume tile ti: rows r0 + ty + 8k (conflict-free LDS reads)
            int r0 = ti * RPT;
            float x[8];
            #pragma unroll
            for (int k = 0; k < 8; ++k) {
                int r = ty + 8 * k;
                int i = r0 + r;
                bool rowv = rowm[b * MM + i] != 0;
                float sv = cur[r * CPB + jx];
                float sc = (rowv && colv) ? sv : NEG_INF;
                x[k] = sc + ub[i];
            }
            float cm = x[0];
            #pragma unroll
            for (int k = 1; k < 8; ++k) cm = fmaxf(cm, x[k]);
            float cs = 0.0f;
            #pragma unroll
            for (int k = 0; k < 8; ++k) cs += expf(x[k] - cm);
            float nm = fmaxf(m, cm);
            s = s * expf(m - nm) + cs * expf(cm - nm);
            m = nm;
            __syncthreads();   // tile ti fully consumed; buffer reusable
        }
        // dustbin row i = MM (no score load); fold in once per column
        if (ty == 0) {
            float x = (colv ? alpha : NEG_INF) + ub[MM];
            float nm = fmaxf(m, x);
            s = s * expf(m - nm) + expf(x - nm);
            m = nm;
        }
    } else {
        // ---- dustbin-column block: only j == NN is live, no score loads ----
        for (int i = ty; i < MP; i += 8) {
            float sc;
            if (i < MM) sc = (rowm[b * MM + i] != 0) ? alpha : NEG_INF;
            else        sc = alpha;
            float x = sc + ub[i];
            float nm = fmaxf(m, x);
            s = s * expf(m - nm) + expf(x - nm);
            m = nm;
        }
    }

    // cross-ty LSE merge of (m, s) pairs per column
    red_m[tt] = m;
    red_s[tt] = s;
    __syncthreads();
    for (int st = 4; st > 0; st >>= 1) {
        if (ty < st) {
            float m1 = red_m[ty * 32 + jx];
            float s1 = red_s[ty * 32 + jx];
            float m2 = red_m[(ty + st) * 32 + jx];
            float s2 = red_s[(ty + st) * 32 + jx];
            float nm = fmaxf(m1, m2);
            red_m[ty * 32 + jx] = nm;
            red_s[ty * 32 + jx] = s1 * expf(m1 - nm) + s2 * expf(m2 - nm);
        }
        __syncthreads();
    }
    if (ty == 0 && j < NP)
        v[b * NP + j] = log_nu[b * NP + j] - (red_m[jx] + logf(red_s[jx]));
}

// ---------------------------------------------------------------------------
// finalize: out = S + u + v - norm   (one block per (b,row), streaming write)
// ---------------------------------------------------------------------------
__global__ __launch_bounds__(256) void sink_final(
    const float* __restrict__ scores, const float* __restrict__ alpha_p,
    const unsigned char* __restrict__ rowm, const unsigned char* __restrict__ colm,
    const float* __restrict__ u, const float* __restrict__ v,
    const float* __restrict__ normb, float* __restrict__ out) {
    int bi = blockIdx.x;
    int b = bi / MP;
    int i = bi % MP;
    int t = threadIdx.x;
    float alpha = alpha_p[0];
    float ui = u[b * MP + i];
    float nrm = normb[b];
    bool rowv = (i < MM) ? (rowm[b * MM + i] != 0) : true;
    const float* rowp = scores + ((size_t)(b * MM + (i < MM ? i : 0))) * NN;
    const float* vb = v + b * NP;
    float* orow = out + ((size_t)b * MP + i) * NP;

    for (int j = t; j < NP; j += 256) {
        float s;
        if (i < MM) {
            if (j < NN) {
                bool colv = colm[b * NN + j] != 0;
                float sv = rowp[j];
                s = (rowv && colv) ? sv : NEG_INF;
            } else {
                s = rowv ? alpha : NEG_INF;
            }
        } else {
            if (j < NN) {
                bool colv = colm[b * NN + j] != 0;
                s = colv ? alpha : NEG_INF;
            } else {
                s = alpha;
            }
        }
        orow[j] = s + ui + vb[j] - nrm;
    }
}

// ---------------------------------------------------------------------------
extern "C" void kernel_launch(void* const* d_in, const int* in_sizes, int n_in,
                              void* d_out, int out_size, void* d_ws, size_t ws_size,
                              hipStream_t stream) {
    const float* scores        = (const float*)d_in[0];
    const float* alpha         = (const float*)d_in[1];
    const unsigned char* rowm  = (const unsigned char*)d_in[2];
    const unsigned char* colm  = (const unsigned char*)d_in[3];
    float* out = (float*)d_out;

    // workspace layout (floats): u, v, log_mu, log_nu, norm  (~262KB)
    float* ws     = (float*)d_ws;
    float* u      = ws;
    float* v      = u + BB * MP;
    float* log_mu = v + BB * NP;
    float* log_nu = log_mu + BB * MP;
    float* normb  = log_nu + BB * NP;

    sink_init<<<BB, 256, 0, stream>>>(rowm, colm, alpha, u, v, log_mu, log_nu, normb);

    for (int it = 0; it < NUM_ITERS; ++it) {
        sink_row<<<BB * MP, 256, 0, stream>>>(scores, alpha, rowm, colm, v, u, log_mu);
        sink_col<<<BB * NJB, 256, 0, stream>>>(scores, alpha, rowm, colm, u, v, log_nu);
    }

    sink_final<<<BB * MP, 256, 0, stream>>>(scores, alpha, rowm, colm, u, v, normb, out);
}